// KeepTop32_35957466202625
// MI455X (gfx1250) — compile-verified
//
#include <hip/hip_runtime.h>
#include <stdint.h>

// ---------------------------------------------------------------------------
// Problem constants (from reference): inputs (64,64,64,128) f32, k = 400.
// ---------------------------------------------------------------------------
#define BATCH       64
#define N_PER_B     (1u << 19)        // 64*64*128 = 524288 elements per batch
#define K_TOP       400u
#define BINS        4096              // histogram over top 12 bits of key
#define CAP         8192              // candidate buffer per batch
#define TPB         256

// hist pass: each block handles 16384 elements (4096 float4), 32 blocks/batch
#define HIST_F4_PER_BLOCK   4096
#define HIST_ITERS          16        // 4096 / 256
#define HIST_BLOCKS         ((BATCH * N_PER_B) / (HIST_F4_PER_BLOCK * 4))

// mask pass: each block handles 8 tiles of 256 float4 = 8192 floats
#define MASK_TILES          8
#define MASK_BLOCKS         ((BATCH * N_PER_B) / (MASK_TILES * TPB * 4))

// gfx1250 async global->LDS path via inline asm (device compile only)
#if defined(__AMDGCN__) && defined(__gfx1250__)
#define HAVE_ASYNC_LDS 1
#else
#define HAVE_ASYNC_LDS 0
#endif

__device__ __forceinline__ uint32_t f32_key(float x) {
    // monotonic mapping: larger float  <=>  larger unsigned key
    uint32_t u = __float_as_uint(x);
    uint32_t m = ((uint32_t)((int32_t)u >> 31)) | 0x80000000u;
    return u ^ m;
}

#if HAVE_ASYNC_LDS
// Async DMA of 16 bytes per lane: global -> LDS, tracked by ASYNCcnt.
__device__ __forceinline__ void async_copy_b128(void* lds_dst, const void* gsrc) {
    uint32_t lds_addr = (uint32_t)(uintptr_t)lds_dst;   // low 32 bits = LDS byte offset
    uint64_t gaddr    = (uint64_t)(uintptr_t)gsrc;
    asm volatile("global_load_async_to_lds_b128 %0, %1, off"
                 :
                 : "v"(lds_addr), "v"(gaddr)
                 : "memory");
}
#endif

// Non-temporal float4 store: output is write-once, never re-read -> keep the
// (L2-resident, 128MB < 192MB L2) input from being evicted by output traffic.
__device__ __forceinline__ void store_nt_f4(float4* dst, float4 v) {
    __builtin_nontemporal_store(v.x, &dst->x);
    __builtin_nontemporal_store(v.y, &dst->y);
    __builtin_nontemporal_store(v.z, &dst->z);
    __builtin_nontemporal_store(v.w, &dst->w);
}

// ---------------------------------------------------------------------------
// Pass 1: per-batch histogram of top-12 key bits (LDS hist -> global atomics)
// ---------------------------------------------------------------------------
__global__ void k_hist(const float4* __restrict__ in, uint32_t* __restrict__ hist) {
    __shared__ uint32_t lh[BINS];
    for (int i = threadIdx.x; i < BINS; i += TPB) lh[i] = 0u;
    __syncthreads();

    const uint32_t batch = blockIdx.x >> 5;                 // 32 blocks / batch
    const uint32_t base4 = blockIdx.x * HIST_F4_PER_BLOCK;

    for (int it = 0; it < HIST_ITERS; ++it) {
        const uint32_t idx4 = base4 + it * TPB + threadIdx.x;
#if defined(__AMDGCN__)
        if (it + 4 < HIST_ITERS)
            __builtin_prefetch(&in[idx4 + 4 * TPB], 0, 0);  // global_prefetch_b8
#endif
        float4 v = in[idx4];
        atomicAdd(&lh[f32_key(v.x) >> 20], 1u);
        atomicAdd(&lh[f32_key(v.y) >> 20], 1u);
        atomicAdd(&lh[f32_key(v.z) >> 20], 1u);
        atomicAdd(&lh[f32_key(v.w) >> 20], 1u);
    }
    __syncthreads();

    uint32_t* gh = hist + batch * BINS;
    for (int i = threadIdx.x; i < BINS; i += TPB) {
        uint32_t c = lh[i];
        if (c) atomicAdd(&gh[i], c);
    }
}

// ---------------------------------------------------------------------------
// Pass 2: find bin containing the K-th largest + residual rank (1 block/batch)
// ---------------------------------------------------------------------------
__global__ void k_findbin(const uint32_t* __restrict__ hist,
                          uint32_t* __restrict__ binIdx,
                          uint32_t* __restrict__ kRem) {
    __shared__ uint32_t sH[BINS];
    __shared__ uint32_t sC[TPB];
    const uint32_t b = blockIdx.x;
    const uint32_t* h = hist + b * BINS;

    for (int i = threadIdx.x; i < BINS; i += TPB) sH[i] = h[i];
    __syncthreads();

    // chunk sums: thread t owns bins [t*16, t*16+16)
    uint32_t s = 0;
    for (int j = 0; j < 16; ++j) s += sH[threadIdx.x * 16 + j];
    sC[threadIdx.x] = s;
    __syncthreads();

    if (threadIdx.x == 0) {
        uint32_t cum = 0;
        for (int ch = TPB - 1; ch >= 0; --ch) {
            uint32_t cs = sC[ch];
            if (cum + cs >= K_TOP) {
                for (int bin = ch * 16 + 15; bin >= ch * 16; --bin) {
                    uint32_t c = sH[bin];
                    if (cum + c >= K_TOP) {
                        binIdx[b] = (uint32_t)bin;
                        kRem[b]   = K_TOP - cum;
                        return;
                    }
                    cum += c;
                }
            }
            cum += cs;
        }
        binIdx[b] = 0u;
        kRem[b]   = 1u;
    }
}

// ---------------------------------------------------------------------------
// Pass 3: compact full keys of elements in the threshold bin
// ---------------------------------------------------------------------------
__global__ void k_compact(const float4* __restrict__ in,
                          const uint32_t* __restrict__ binIdx,
                          uint32_t* __restrict__ cnt,
                          uint32_t* __restrict__ cand) {
    const uint32_t batch  = blockIdx.x >> 5;
    const uint32_t target = binIdx[batch];
    const uint32_t base4  = blockIdx.x * HIST_F4_PER_BLOCK;
    uint32_t* cb = cand + batch * CAP;

    for (int it = 0; it < HIST_ITERS; ++it) {
        const uint32_t idx4 = base4 + it * TPB + threadIdx.x;
        float4 v = in[idx4];
        uint32_t k0 = f32_key(v.x), k1 = f32_key(v.y);
        uint32_t k2 = f32_key(v.z), k3 = f32_key(v.w);
        if ((k0 >> 20) == target) { uint32_t p = atomicAdd(&cnt[batch], 1u); if (p < CAP) cb[p] = k0; }
        if ((k1 >> 20) == target) { uint32_t p = atomicAdd(&cnt[batch], 1u); if (p < CAP) cb[p] = k1; }
        if ((k2 >> 20) == target) { uint32_t p = atomicAdd(&cnt[batch], 1u); if (p < CAP) cb[p] = k2; }
        if ((k3 >> 20) == target) { uint32_t p = atomicAdd(&cnt[batch], 1u); if (p < CAP) cb[p] = k3; }
    }
}

// ---------------------------------------------------------------------------
// Pass 4: exact threshold key via binary search on low 20 bits (1 block/batch)
// ---------------------------------------------------------------------------
__global__ void k_select(const uint32_t* __restrict__ cand,
                         const uint32_t* __restrict__ cnt,
                         const uint32_t* __restrict__ binIdx,
                         const uint32_t* __restrict__ kRem,
                         uint32_t* __restrict__ thr) {
    __shared__ uint32_t sCand[CAP];
    __shared__ uint32_t sCount;
    const uint32_t b = blockIdx.x;
    const uint32_t C = min(cnt[b], (uint32_t)CAP);
    const uint32_t kr = kRem[b];
    const uint32_t* src = cand + b * CAP;

    for (uint32_t i = threadIdx.x; i < C; i += TPB) sCand[i] = src[i] & 0xFFFFFu;
    __syncthreads();

    uint32_t lo = 0u, hi = 0xFFFFFu;
    while (lo < hi) {                       // uniform across the block
        uint32_t mid = lo + ((hi - lo + 1u) >> 1);
        if (threadIdx.x == 0) sCount = 0u;
        __syncthreads();
        uint32_t p = 0;
        for (uint32_t i = threadIdx.x; i < C; i += TPB) p += (sCand[i] >= mid) ? 1u : 0u;
        if (p) atomicAdd(&sCount, p);
        __syncthreads();
        uint32_t total = sCount;
        __syncthreads();
        if (total >= kr) lo = mid; else hi = mid - 1u;
    }
    if (threadIdx.x == 0) thr[b] = (binIdx[b] << 20) | lo;
}

// ---------------------------------------------------------------------------
// Pass 5: apply mask, streaming through LDS with gfx1250 async copies;
//         non-temporal stores keep the input L2-resident.
// ---------------------------------------------------------------------------
__global__ void k_mask(const float4* __restrict__ in,
                       float4* __restrict__ out,
                       const uint32_t* __restrict__ thr) {
    const uint32_t T = thr[blockIdx.x >> 6];     // 64 blocks per batch
    const uint32_t base4 = blockIdx.x * (MASK_TILES * TPB);

#if HAVE_ASYNC_LDS
    __shared__ float4 stage[2][TPB];
    // issue tile 0
    async_copy_b128(&stage[0][threadIdx.x], &in[base4 + threadIdx.x]);
    for (int t = 0; t < MASK_TILES; ++t) {
        if (t + 1 < MASK_TILES) {
            async_copy_b128(&stage[(t + 1) & 1][threadIdx.x],
                            &in[base4 + (t + 1) * TPB + threadIdx.x]);
            asm volatile("s_wait_asynccnt 1" ::: "memory");
        } else {
            asm volatile("s_wait_asynccnt 0" ::: "memory");
        }
        float4 v = stage[t & 1][threadIdx.x];
        v.x = (f32_key(v.x) >= T) ? v.x : 0.0f;
        v.y = (f32_key(v.y) >= T) ? v.y : 0.0f;
        v.z = (f32_key(v.z) >= T) ? v.z : 0.0f;
        v.w = (f32_key(v.w) >= T) ? v.w : 0.0f;
        store_nt_f4(&out[base4 + t * TPB + threadIdx.x], v);
    }
#else
    for (int t = 0; t < MASK_TILES; ++t) {
        const uint32_t idx4 = base4 + t * TPB + threadIdx.x;
        float4 v = in[idx4];
        v.x = (f32_key(v.x) >= T) ? v.x : 0.0f;
        v.y = (f32_key(v.y) >= T) ? v.y : 0.0f;
        v.z = (f32_key(v.z) >= T) ? v.z : 0.0f;
        v.w = (f32_key(v.w) >= T) ? v.w : 0.0f;
        store_nt_f4(&out[idx4], v);
    }
#endif
}

// ---------------------------------------------------------------------------
// Launch
// ---------------------------------------------------------------------------
extern "C" void kernel_launch(void* const* d_in, const int* in_sizes, int n_in,
                              void* d_out, int out_size, void* d_ws, size_t ws_size,
                              hipStream_t stream) {
    (void)in_sizes; (void)n_in; (void)out_size; (void)ws_size;

    const float4* in  = (const float4*)d_in[0];
    float4*       out = (float4*)d_out;

    // workspace layout (uint32 units)
    uint32_t* ws      = (uint32_t*)d_ws;
    uint32_t* hist    = ws;                         // BATCH*BINS
    uint32_t* binIdx  = hist   + BATCH * BINS;      // BATCH
    uint32_t* kRem    = binIdx + BATCH;             // BATCH
    uint32_t* cnt     = kRem   + BATCH;             // BATCH
    uint32_t* thr     = cnt    + BATCH;             // BATCH
    uint32_t* cand    = thr    + BATCH;             // BATCH*CAP

    // zero histogram + the four small arrays (contiguous at front of ws)
    const size_t zero_bytes = (size_t)(BATCH * BINS + 4 * BATCH) * sizeof(uint32_t);
    (void)hipMemsetAsync(d_ws, 0, zero_bytes, stream);

    k_hist   <<<HIST_BLOCKS, TPB, 0, stream>>>(in, hist);
    k_findbin<<<BATCH,       TPB, 0, stream>>>(hist, binIdx, kRem);
    k_compact<<<HIST_BLOCKS, TPB, 0, stream>>>(in, binIdx, cnt, cand);
    k_select <<<BATCH,       TPB, 0, stream>>>(cand, cnt, binIdx, kRem, thr);
    k_mask   <<<MASK_BLOCKS, TPB, 0, stream>>>(in, out, thr);
}